// MS_AFDS_31696858644715
// MI455X (gfx1250) — compile-verified
//
#include <hip/hip_runtime.h>
#include <math.h>

#define NB 97
#define NBP 112          // 7 * 16 (padded for WMMA tiles)
#define DD 512
#define SPLIT 16

typedef float v2f __attribute__((ext_vector_type(2)));
typedef float v8f __attribute__((ext_vector_type(8)));

struct SmoothKern { float w[15]; }; // w3 @0 (3), w5 @3 (5), w7 @8 (7)

// ---------------- host: replicate scipy gaussian_filter1d window ----------------
static void kernel_window_host(int ks, double sigma, float* out) {
    int half = (ks - 1) / 2;
    int r = (int)(4.0 * sigma + 0.5);
    int glen = 2 * r + 1;
    double g[32];
    double gs = 0.0;
    for (int m = 0; m < glen; ++m) {
        double t = (double)(m - r);
        g[m] = exp(-0.5 * t * t / (sigma * sigma));
        gs += g[m];
    }
    for (int m = 0; m < glen; ++m) g[m] /= gs;
    double base[8] = {0};
    base[half] = 1.0;
    int plen = ks + 2 * r;
    double xp[40];
    int period = 2 * ks;
    for (int p = 0; p < plen; ++p) {               // numpy 'symmetric' pad
        int q = p - r;
        int qq = ((q % period) + period) % period;
        int src = (qq < ks) ? qq : (period - 1 - qq);
        xp[p] = base[src];
    }
    double w[8];
    double ws = 0.0;
    for (int i = 0; i < ks; ++i) {                  // np.convolve 'valid'
        double acc = 0.0;
        for (int m = 0; m < glen; ++m) acc += xp[i + m] * g[glen - 1 - m];
        w[i] = acc; ws += acc;
    }
    for (int i = 0; i < ks; ++i) out[i] = (float)(w[i] / ws);
}

// ---------------- K1: bucketize + histogram ----------------
__global__ void hist_kernel(const int* __restrict__ labels, int* __restrict__ bkt,
                            int* __restrict__ cnt) {
    __shared__ int h[NB];
    int t = threadIdx.x;
    if (t < NB) h[t] = 0;
    __syncthreads();
    int i = blockIdx.x * 256 + t;
    int lab = labels[i];
    int b = min(max(lab, 3), 99) - 3;
    bkt[i] = b;
    atomicAdd(&h[b], 1);
    __syncthreads();
    if (t < NB && h[t]) atomicAdd(&cnt[t], h[t]);
}

// ---------------- K2: exclusive scan over 97 counts ----------------
__global__ void scan_kernel(const int* __restrict__ cnt, int* __restrict__ off,
                            int* __restrict__ cur) {
    if (threadIdx.x == 0) {
        int r = 0;
        for (int b = 0; b < NB; ++b) { off[b] = r; cur[b] = r; r += cnt[b]; }
    }
}

// ---------------- K3: scatter sample indices per bucket ----------------
__global__ void scatter_kernel(const int* __restrict__ bkt, int* __restrict__ cur,
                               int* __restrict__ idx) {
    int i = blockIdx.x * 256 + threadIdx.x;
    int b = bkt[i];
    int pos = atomicAdd(&cur[b], 1);
    idx[pos] = i;
}

// ---------------- K4: segment sums (register accumulation, coalesced rows) ----------------
__global__ void segsum_kernel(const float* __restrict__ feat, const int* __restrict__ idx,
                              const int* __restrict__ cnt, const int* __restrict__ off,
                              float* __restrict__ s, float* __restrict__ sq) {
    int b = blockIdx.x;
    int c = cnt[b], st = off[b];
    int chunk = (c + SPLIT - 1) / SPLIT;
    int j0 = blockIdx.y * chunk;
    int j1 = min(j0 + chunk, c);
    int d = threadIdx.x;                    // 512 threads = 512 columns
    float sum = 0.f, ssq = 0.f;
    for (int j = j0; j < j1; ++j) {
        int i = idx[st + j];
        float v = feat[i * DD + d];
        sum += v;
        ssq = fmaf(v, v, ssq);
    }
    if (j1 > j0) {
        atomicAdd(&s[b * DD + d], sum);
        atomicAdd(&sq[b * DD + d], ssq);
    }
}

// ---------------- K5: mean/var -> EMA-updated rm/rv (padded to 112 rows) ----------------
__global__ void stats_kernel(const float* __restrict__ rmean_in, const float* __restrict__ rvar_in,
                             const float* __restrict__ s, const float* __restrict__ sq,
                             const int* __restrict__ cnt, const int* __restrict__ epoch_p,
                             float* __restrict__ rm_p, float* __restrict__ rv_p) {
    int e = blockIdx.x * 256 + threadIdx.x;      // 0 .. NBP*DD-1
    int r = e >> 9;
    if (r >= NB) { rm_p[e] = 0.f; rv_p[e] = 0.f; return; }
    float n = (float)cnt[r];
    float nc = fmaxf(n, 1.f);
    float mean = s[e] / nc;
    float biased = fmaxf(sq[e] / nc - mean * mean, 0.f);
    float denom = (n > 1.f) ? (n - 1.f) : nc;
    float var = biased * (n / denom);
    float factor = (epoch_p[0] != 0) ? 0.9f : 0.f;
    float rm0 = rmean_in[e], rv0 = rvar_in[e];
    bool present = (n > 0.f);
    rm_p[e] = present ? ((1.f - factor) * mean + factor * rm0) : rm0;
    rv_p[e] = present ? ((1.f - factor) * var  + factor * rv0) : rv0;
}

// ---------------- K6: build combined reflect-pad conv operator W [112x112] ----------------
__global__ void buildW_kernel(SmoothKern kw, float* __restrict__ W) {
    int i = threadIdx.x;                      // 0..111
    float* row = W + i * NBP;
    for (int j = 0; j < NBP; ++j) row[j] = 0.f;
    if (i < NB) {
        const int ks[3]   = {3, 5, 7};
        const int offs[3] = {0, 3, 8};
        for (int k = 0; k < 3; ++k) {
            int h = (ks[k] - 1) / 2;
            for (int t = 0; t < ks[k]; ++t) {
                int q = i + t - h;                               // reflect (no edge repeat)
                int src = (q < 0) ? -q : ((q > NB - 1) ? 2 * (NB - 1) - q : q);
                row[src] += kw.w[offs[k] + t] * (1.f / 3.f);
            }
        }
    }
}

// ---------------- K7: alpha[b] = exp(-nst / (mean(nst)+eps)) ----------------
__global__ void alpha_kernel(const float* __restrict__ nst_in, const int* __restrict__ cnt,
                             float* __restrict__ alpha) {
    if (threadIdx.x == 0) {
        float sum = 0.f;
        for (int b = 0; b < NB; ++b) sum += nst_in[b] + (float)cnt[b];
        float mean = sum / (float)NB;
        for (int b = 0; b < NB; ++b)
            alpha[b] = expf(-(nst_in[b] + (float)cnt[b]) / (mean + 1e-5f));
    }
}

// ---------------- K8: smoothing as fp32 WMMA GEMM  sm = W(112x112) x rm_p(112x512) ----------------
// one 16x16 output tile per wave; M-tiles=7, N-tiles=32, K covered 0..99 in steps of 4
__global__ void smooth_wmma_kernel(const float* __restrict__ W,
                                   const float* __restrict__ rm_p, const float* __restrict__ rv_p,
                                   float* __restrict__ smm, float* __restrict__ smv) {
    int wave = blockIdx.x * 8 + (threadIdx.x >> 5);   // 448 waves total
    int lane = threadIdx.x & 31;
    int q  = wave / 224;                              // 0: mean, 1: var
    int r  = wave % 224;
    int mt = r / 32, nt = r % 32;
    const float* src = q ? rv_p : rm_p;
    float*       dst = q ? smv  : smm;
    int ln = lane & 15;
    int hi = lane >> 4;
    int col = nt * 16 + ln;
    const float* Wrow = W + (mt * 16 + ln) * NBP;
#if __has_builtin(__builtin_amdgcn_wmma_f32_16x16x4_f32)
    v8f acc = {0.f, 0.f, 0.f, 0.f, 0.f, 0.f, 0.f, 0.f};
    for (int kk = 0; kk < 25; ++kk) {
        int k0 = kk * 4;
        // A 16x4 f32 layout: VGPR0 = {K0 | K2}, VGPR1 = {K1 | K3} split at lane16
        v2f a, b;
        a.x = Wrow[k0 + 2 * hi];
        a.y = Wrow[k0 + 2 * hi + 1];
        // B 4x16 f32: row striped across lanes within a VGPR, same K split
        b.x = src[(k0 + 2 * hi) * DD + col];
        b.y = src[(k0 + 2 * hi + 1) * DD + col];
        acc = __builtin_amdgcn_wmma_f32_16x16x4_f32(
            false, a, false, b, (short)0, acc, false, false);
    }
    #pragma unroll
    for (int rr = 0; rr < 8; ++rr) {
        int rowm = mt * 16 + rr + hi * 8;             // C/D layout: VGPR rr -> M=rr / rr+8
        dst[rowm * DD + col] = acc[rr];
    }
#else
    // scalar fallback (same tile assignment)
    for (int rr = 0; rr < 8; ++rr) {
        int rowm = mt * 16 + rr + hi * 8;
        const float* wr = W + rowm * NBP;
        float acc = 0.f;
        for (int k = 0; k < NB; ++k) acc = fmaf(wr[k], src[k * DD + col], acc);
        dst[rowm * DD + col] = acc;
    }
#endif
}

// ---------------- K9: fold everything into per-(bucket,col) scale/offset ----------------
__global__ void mix_kernel(const float* __restrict__ rm_p, const float* __restrict__ rv_p,
                           const float* __restrict__ smm, const float* __restrict__ smv,
                           const float* __restrict__ alpha,
                           float* __restrict__ scA, float* __restrict__ offA) {
    int e = blockIdx.x * 256 + threadIdx.x;           // < NB*DD
    int r = e >> 9;
    float al = alpha[r];
    float m1 = rm_p[e], v1 = rv_p[e];
    float m2 = (1.f - al) * m1 + al * smm[e];
    float v2 = (1.f - al) * v1 + al * smv[e];
    float sc, of;
    if (v1 > 0.f) {
        float ratio = v2 / fmaxf(v1, 1e-30f);
        ratio = fminf(fmaxf(ratio, 0.1f), 10.f);
        sc = sqrtf(ratio);
        of = m2 - m1 * sc;                            // (f-m1)*sc+m2 == f*sc + of
    } else { sc = 1.f; of = 0.f; }                    // identity branch
    scA[e] = sc; offA[e] = of;
}

// ---------------- K10: apply, one fma per element, float4 vectorized ----------------
__global__ void apply_kernel(const float4* __restrict__ f4, const int* __restrict__ bkt,
                             const float4* __restrict__ sc4, const float4* __restrict__ of4,
                             float4* __restrict__ out4) {
    unsigned g = blockIdx.x * 256 + threadIdx.x;      // N*D/4 = 16.7M
    int row = (int)(g >> 7);                          // 128 float4 per row
    int c4  = (int)(g & 127);
    int b = bkt[row];
    float4 f = f4[g];
    float4 s = sc4[b * 128 + c4];
    float4 o = of4[b * 128 + c4];
    float4 r;
    r.x = fmaf(f.x, s.x, o.x);
    r.y = fmaf(f.y, s.y, o.y);
    r.z = fmaf(f.z, s.z, o.z);
    r.w = fmaf(f.w, s.w, o.w);
    out4[g] = r;
}

extern "C" void kernel_launch(void* const* d_in, const int* in_sizes, int n_in,
                              void* d_out, int out_size, void* d_ws, size_t ws_size,
                              hipStream_t stream) {
    const float* features = (const float*)d_in[0];
    const int*   labels   = (const int*)d_in[1];
    const float* rmean_in = (const float*)d_in[2];
    const float* rvar_in  = (const float*)d_in[3];
    const float* nst_in   = (const float*)d_in[4];
    const int*   epoch_p  = (const int*)d_in[5];
    const int N = in_sizes[0] / DD;                   // 131072

    // ---- carve workspace (s, sq, cnt first so one memset clears them) ----
    char* p = (char*)d_ws;
    auto carve = [&](size_t bytes) { void* r = (void*)p; p += (bytes + 255) & ~(size_t)255; return r; };
    float* s_buf  = (float*)carve(NB * DD * sizeof(float));
    float* sq_buf = (float*)carve(NB * DD * sizeof(float));
    int*   cnt    = (int*)  carve(NB * sizeof(int));
    size_t zero_bytes = (size_t)(p - (char*)d_ws);
    int*   off    = (int*)  carve(NB * sizeof(int));
    int*   cur    = (int*)  carve(NB * sizeof(int));
    int*   bkt    = (int*)  carve((size_t)N * sizeof(int));
    int*   idx    = (int*)  carve((size_t)N * sizeof(int));
    float* rm_p   = (float*)carve(NBP * DD * sizeof(float));
    float* rv_p   = (float*)carve(NBP * DD * sizeof(float));
    float* smm    = (float*)carve(NBP * DD * sizeof(float));
    float* smv    = (float*)carve(NBP * DD * sizeof(float));
    float* Wop    = (float*)carve(NBP * NBP * sizeof(float));
    float* alpha  = (float*)carve(NB * sizeof(float));
    float* scA    = (float*)carve(NB * DD * sizeof(float));
    float* offA   = (float*)carve(NB * DD * sizeof(float));

    // host: gaussian windows (exact double-precision replication of reference)
    SmoothKern kw;
    kernel_window_host(3, 1.0, kw.w + 0);
    kernel_window_host(5, 2.0, kw.w + 3);
    kernel_window_host(7, 3.0, kw.w + 8);

    hipMemsetAsync(d_ws, 0, zero_bytes, stream);

    hist_kernel   <<<N / 256, 256, 0, stream>>>(labels, bkt, cnt);
    scan_kernel   <<<1, 32, 0, stream>>>(cnt, off, cur);
    scatter_kernel<<<N / 256, 256, 0, stream>>>(bkt, cur, idx);
    segsum_kernel <<<dim3(NB, SPLIT), DD, 0, stream>>>(features, idx, cnt, off, s_buf, sq_buf);
    stats_kernel  <<<(NBP * DD) / 256, 256, 0, stream>>>(rmean_in, rvar_in, s_buf, sq_buf,
                                                         cnt, epoch_p, rm_p, rv_p);
    buildW_kernel <<<1, NBP, 0, stream>>>(kw, Wop);
    alpha_kernel  <<<1, 32, 0, stream>>>(nst_in, cnt, alpha);
    smooth_wmma_kernel<<<56, 256, 0, stream>>>(Wop, rm_p, rv_p, smm, smv);
    mix_kernel    <<<(NB * DD) / 256, 256, 0, stream>>>(rm_p, rv_p, smm, smv, alpha, scA, offA);
    apply_kernel  <<<((unsigned)N * DD / 4) / 256, 256, 0, stream>>>(
        (const float4*)features, bkt, (const float4*)scA, (const float4*)offA, (float4*)d_out);
}